// GCN_19310172963198
// MI455X (gfx1250) — compile-verified
//
#include <hip/hip_runtime.h>

// ---------------------------------------------------------------------------
// 2-layer GCN for MI455X (gfx1250, wave32).
//   out = GCN(relu(GCN(x, W1, b1)), W2, b2)  with  A_hat = D^-1/2 (A+I) D^-1/2
// Dense transforms: V_WMMA_F32_16X16X4_F32 (exact fp32 matrix pipe), with the
// A tile staged once per block into LDS via GLOBAL_LOAD_ASYNC_TO_LDS_B128
// (ASYNCcnt-tracked DMA path), read back via ds_load_b64 in the documented
// 16x4 fp32 fragment layout.
// Sparse aggregation: L2-resident float4 gathers + global_atomic_add_f32
// (feature tables fit in the 192 MB L2 -> L2-BW bound, not HBM bound).
// ---------------------------------------------------------------------------

typedef __attribute__((ext_vector_type(2))) float v2f;
typedef __attribute__((ext_vector_type(8))) float v8f;
typedef __attribute__((ext_vector_type(4))) int   v4i;

#define F_IN 64
#define HID  128
#define NCLS 32

#if __has_builtin(__builtin_amdgcn_global_load_async_to_lds_b128)
#define HAVE_ASYNC_LDS 1
typedef __attribute__((address_space(1))) v4i* g_v4i_p;   // global
typedef __attribute__((address_space(3))) v4i* l_v4i_p;   // LDS
#else
#define HAVE_ASYNC_LDS 0
#endif

// ---------------- degree / normalization ----------------
__global__ void k_init_deg(float* deg, int n) {
    int i = blockIdx.x * blockDim.x + threadIdx.x;
    if (i < n) deg[i] = 1.0f;                      // self loop
}

__global__ void k_count_deg(const int* __restrict__ dst, float* deg, int e) {
    int i = blockIdx.x * blockDim.x + threadIdx.x;
    if (i < e) atomicAdd(&deg[dst[i]], 1.0f);
}

__global__ void k_dinv(float* deg, int n) {
    int i = blockIdx.x * blockDim.x + threadIdx.x;
    if (i < n) deg[i] = rsqrtf(fmaxf(deg[i], 1.0f));   // in-place: deg -> dinv
}

// ---------------- dense GEMM via WMMA fp32, LDS-staged A ----------------
// C[M x NC] = A[M x K] * B[K x NC], row-major. M % 16 == 0.
// One block = NC/16 waves; block b computes rows [16b, 16b+16) x all NC cols.
// The 16xK A tile is loaded to LDS once (async DMA path if available) and
// shared by all waves; B (tiny weight matrix) streams from L2/WGP$.
// WMMA fp32 fragment layout: lanes 0-15 hold M=lane, K={k0,k0+1};
// lanes 16-31 hold M=lane-16, K={k0+2,k0+3}. C/D: VGPR r -> M=r(+8 hi half).
template <int K, int NC>
__global__ __launch_bounds__(32 * (NC / 16)) void k_gemm_wmma(
    const float* __restrict__ A, const float* __restrict__ B,
    float* __restrict__ C)
{
    constexpr int WAVES = NC / 16;
    constexpr int LROW  = K + 4;                 // pad: bank spread + 16B align
    __shared__ float As[16 * LROW];

    const int tid  = threadIdx.x;
    const int lane = tid & 31;
    const int wave = tid >> 5;
    const int trow = blockIdx.x;

    // ---- stage 16xK A tile into LDS (one float4 chunk per step) ----
    constexpr int CH = (16 * K) / 4;
    for (int t = tid; t < CH; t += 32 * WAVES) {
        const int r  = t / (K / 4);
        const int c4 = (t - r * (K / 4)) * 4;
        const float* g = A + (size_t)(trow * 16 + r) * K + c4;
        float*       l = As + r * LROW + c4;
#if HAVE_ASYNC_LDS
        __builtin_amdgcn_global_load_async_to_lds_b128(
            (g_v4i_p)g, (l_v4i_p)l, 0, 0);
#else
        *(float4*)l = *(const float4*)g;
#endif
    }
#if HAVE_ASYNC_LDS
#if __has_builtin(__builtin_amdgcn_s_wait_asynccnt)
    __builtin_amdgcn_s_wait_asynccnt(0);
#else
    asm volatile("s_wait_asynccnt 0x0" ::: "memory");
#endif
#endif
    __syncthreads();

    // ---- per-wave 16x16 tile: col block = wave ----
    const int m16 = lane & 15;
    const int hi  = lane >> 4;
    const int col = wave * 16 + m16;

    v8f acc = {};
#pragma unroll
    for (int k0 = 0; k0 < K; k0 += 4) {
        const int ka = k0 + hi * 2;
        float2 av = *(const float2*)(As + m16 * LROW + ka);   // ds_load_b64
        v2f a; a.x = av.x; a.y = av.y;
        v2f b;
        b.x = B[(size_t)ka       * NC + col];
        b.y = B[(size_t)(ka + 1) * NC + col];
        acc = __builtin_amdgcn_wmma_f32_16x16x4_f32(
            false, a, false, b, (short)0, acc, false, false);
    }

    const int rbase = trow * 16 + hi * 8;
#pragma unroll
    for (int r = 0; r < 8; ++r)
        C[(size_t)(rbase + r) * NC + col] = acc[r];
}

// ---------------- self-loop + bias init:  out[i][j] = b[j] + h[i][j]*dinv[i]^2
template <int F>
__global__ __launch_bounds__(256) void k_selfloop_bias(
    const float* __restrict__ h, const float* __restrict__ dinv,
    const float* __restrict__ bias, float* __restrict__ out, int n)
{
    int idx = blockIdx.x * blockDim.x + threadIdx.x;
    if (idx >= n * F) return;
    int i = idx / F;
    int j = idx - i * F;
    float di = dinv[i];
    out[idx] = bias[j] + h[idx] * di * di;
}

// ---------------- edge scatter:  out[dst] += h[src] * dinv[src]*dinv[dst]
// Thread t handles one float4 chunk of one edge; a wave covers F/4 lanes per
// edge so gathers of h[src] are coalesced bursts (tables are L2-resident).
template <int F>
__global__ __launch_bounds__(256) void k_edge_scatter(
    const float* __restrict__ h, const float* __restrict__ dinv,
    const int* __restrict__ src, const int* __restrict__ dst,
    float* __restrict__ out, int e)
{
    constexpr int CH = F / 4;                         // float4 chunks per edge
    long long gid = (long long)blockIdx.x * blockDim.x + threadIdx.x;
    if (gid >= (long long)e * CH) return;
    int eidx = (int)(gid / CH);
    int c    = (int)(gid - (long long)eidx * CH);
    int s = src[eidx];
    int d = dst[eidx];
    float w = dinv[s] * dinv[d];
    float4 v = *(const float4*)(h + (size_t)s * F + c * 4);
    float* o = out + (size_t)d * F + c * 4;
    atomicAdd(o + 0, v.x * w);
    atomicAdd(o + 1, v.y * w);
    atomicAdd(o + 2, v.z * w);
    atomicAdd(o + 3, v.w * w);
}

// ---------------- ReLU ----------------
__global__ void k_relu(float* x, int n) {
    int i = blockIdx.x * blockDim.x + threadIdx.x;
    if (i < n) x[i] = fmaxf(x[i], 0.0f);
}

// ---------------------------------------------------------------------------
extern "C" void kernel_launch(void* const* d_in, const int* in_sizes, int n_in,
                              void* d_out, int out_size, void* d_ws, size_t ws_size,
                              hipStream_t stream) {
    (void)n_in; (void)out_size; (void)ws_size;

    const float* x   = (const float*)d_in[0];
    const int*   ei  = (const int*)  d_in[1];
    const float* W1  = (const float*)d_in[2];
    const float* b1  = (const float*)d_in[3];
    const float* W2  = (const float*)d_in[4];
    const float* b2  = (const float*)d_in[5];
    float*       out = (float*)d_out;

    const int n = in_sizes[0] / F_IN;     // 100000 (multiple of 16)
    const int e = in_sizes[1] / 2;        // 1600000
    const int* src = ei;                  // edge_index[0]
    const int* dst = ei + e;              // edge_index[1]

    // workspace layout (fp32): dinv[n] | h1[n*HID] | agg1[n*HID] | h2[n*NCLS]
    float* dinv = (float*)d_ws;
    float* h1   = dinv + n;
    float* agg1 = h1 + (size_t)n * HID;
    float* h2   = agg1 + (size_t)n * HID;

    const int T = 256;

    // degrees -> dinv (shared by both layers; graph is identical)
    k_init_deg<<<(n + T - 1) / T, T, 0, stream>>>(dinv, n);
    k_count_deg<<<(e + T - 1) / T, T, 0, stream>>>(dst, dinv, e);
    k_dinv<<<(n + T - 1) / T, T, 0, stream>>>(dinv, n);

    // ---- layer 1 ----
    // h1 = x @ W1   (WMMA fp32; 8 waves/block, A tile LDS-staged once)
    k_gemm_wmma<F_IN, HID><<<n / 16, 32 * (HID / 16), 0, stream>>>(x, W1, h1);
    // agg1 = b1 + selfloop + edge scatter; then ReLU
    k_selfloop_bias<HID><<<((size_t)n * HID + T - 1) / T, T, 0, stream>>>(h1, dinv, b1, agg1, n);
    {
        long long total = (long long)e * (HID / 4);
        k_edge_scatter<HID><<<(unsigned)((total + T - 1) / T), T, 0, stream>>>(h1, dinv, src, dst, agg1, e);
    }
    k_relu<<<((size_t)n * HID + T - 1) / T, T, 0, stream>>>(agg1, n * HID);

    // ---- layer 2 ----
    // h2 = relu(agg1) @ W2   (2 waves/block)
    k_gemm_wmma<HID, NCLS><<<n / 16, 32 * (NCLS / 16), 0, stream>>>(agg1, W2, h2);
    // out = b2 + selfloop + edge scatter
    k_selfloop_bias<NCLS><<<((size_t)n * NCLS + T - 1) / T, T, 0, stream>>>(h2, dinv, b2, out, n);
    {
        long long total = (long long)e * (NCLS / 4);
        k_edge_scatter<NCLS><<<(unsigned)((total + T - 1) / T), T, 0, stream>>>(h2, dinv, src, dst, out, e);
    }
}